// GraphAutoencoder_59330678227392
// MI455X (gfx1250) — compile-verified
//
#include <hip/hip_runtime.h>
#include <hip/hip_bf16.h>

typedef __attribute__((ext_vector_type(2))) float v2f;
typedef __attribute__((ext_vector_type(8))) float v8f;

#define NEG_ATT 0.2f
#define NEG_ACT 0.01f

// ---------- helpers ----------
__device__ __forceinline__ unsigned fenc(float f) {
    unsigned u = __float_as_uint(f);
    return (u & 0x80000000u) ? ~u : (u | 0x80000000u);
}
__device__ __forceinline__ float fdec(unsigned u) {
    u = (u & 0x80000000u) ? (u & 0x7fffffffu) : ~u;
    return __uint_as_float(u);
}

__global__ void fill_u32_kernel(unsigned* __restrict__ p, unsigned v, long long n) {
    long long i = (long long)blockIdx.x * blockDim.x + threadIdx.x;
    if (i < n) p[i] = v;
}

// ---------- generic fp32 WMMA GEMM: C[M,N] = A[M,K] @ B[K,N] ----------
// wave = one 16x16 output tile; 8 waves / 256-thread block.
__global__ void gemm_wmma_f32(const float* __restrict__ A, const float* __restrict__ B,
                              float* __restrict__ C, int M, int N, int K) {
    const int lane = threadIdx.x & 31;
    const int wave = threadIdx.x >> 5;
    const int tilesN = N >> 4;
    const long long tilesM = (long long)(M + 15) >> 4;
    long long wid = (long long)blockIdx.x * (blockDim.x >> 5) + wave;
    if (wid >= tilesM * tilesN) return;          // wave-uniform: EXEC stays all-ones
    const int tn = (int)(wid % tilesN);
    const long long tm = wid / tilesN;
    const int m0 = (int)(tm << 4), n0 = tn << 4;

    const int row = lane & 15;                   // A-frag: M index
    const int kp  = (lane >> 4) << 1;            // 0 or 2 (K sub-pair)
    int arow = m0 + row; if (arow >= M) arow = M - 1;
    const int col = n0 + (lane & 15);            // B/C/D: N index

    v8f acc = {};
    for (int k0 = 0; k0 < K; k0 += 4) {
        v2f a, b;
        const float* ap = A + (size_t)arow * K + k0 + kp;
        a[0] = ap[0];
        a[1] = ap[1];
        b[0] = B[(size_t)(k0 + kp) * N + col];
        b[1] = B[(size_t)(k0 + kp + 1) * N + col];
        acc = __builtin_amdgcn_wmma_f32_16x16x4_f32(false, a, false, b,
                                                    (short)0, acc, false, false);
    }
    const int mbase = m0 + ((lane >> 4) << 3);   // C/D layout: VGPR g -> row g (lo half), 8+g (hi half)
    for (int g = 0; g < 8; ++g) {
        int r = mbase + g;
        if (r < M) C[(size_t)r * N + col] = acc[g];
    }
}

// ---------- attention scores: as[n,h] = <h[n,h,:], att_src[h,:]> ----------
__global__ void attn_scores_kernel(const float* __restrict__ h,
                                   const float* __restrict__ att_s,
                                   const float* __restrict__ att_d,
                                   float* __restrict__ as, float* __restrict__ ad,
                                   int N, int H, int C) {
    long long i = (long long)blockIdx.x * blockDim.x + threadIdx.x;
    if (i >= (long long)N * H) return;
    int n = (int)(i / H), hh = (int)(i % H);
    const float* hp = h + (size_t)n * H * C + (size_t)hh * C;
    const float* sp = att_s + (size_t)hh * C;
    const float* dp = att_d + (size_t)hh * C;
    float s = 0.f, d = 0.f;
    for (int c = 0; c < C; ++c) { float v = hp[c]; s += v * sp[c]; d += v * dp[c]; }
    as[i] = s; ad[i] = d;
}

// ---------- edge pass 1: segment max (order-preserving u32 atomicMax) ----------
__global__ void edge_max_kernel(const int* __restrict__ esrc, const int* __restrict__ edst,
                                long long E, int N,
                                const float* __restrict__ as, const float* __restrict__ ad,
                                unsigned* __restrict__ m, int H) {
    long long i = (long long)blockIdx.x * blockDim.x + threadIdx.x;
    if (i >= E + N) return;
    int s = (i < E) ? esrc[i] : (int)(i - E);
    int d = (i < E) ? edst[i] : (int)(i - E);
    for (int hh = 0; hh < H; ++hh) {
        float e = as[(size_t)s * H + hh] + ad[(size_t)d * H + hh];
        e = e > 0.f ? e : NEG_ATT * e;
        atomicMax(&m[(size_t)d * H + hh], fenc(e));
    }
}

// ---------- edge pass 2: segment sum of exp(e - m) ----------
__global__ void edge_sum_kernel(const int* __restrict__ esrc, const int* __restrict__ edst,
                                long long E, int N,
                                const float* __restrict__ as, const float* __restrict__ ad,
                                const unsigned* __restrict__ m, float* __restrict__ ssum,
                                int H) {
    long long i = (long long)blockIdx.x * blockDim.x + threadIdx.x;
    if (i >= E + N) return;
    int s = (i < E) ? esrc[i] : (int)(i - E);
    int d = (i < E) ? edst[i] : (int)(i - E);
    for (int hh = 0; hh < H; ++hh) {
        float e = as[(size_t)s * H + hh] + ad[(size_t)d * H + hh];
        e = e > 0.f ? e : NEG_ATT * e;
        atomicAdd(&ssum[(size_t)d * H + hh], expf(e - fdec(m[(size_t)d * H + hh])));
    }
}

// ---------- edge pass 3: out[dst] += alpha * h[src]  (wave per edge) ----------
__global__ void edge_aggregate_kernel(const int* __restrict__ esrc, const int* __restrict__ edst,
                                      long long E, int N,
                                      const float* __restrict__ as, const float* __restrict__ ad,
                                      const unsigned* __restrict__ m, const float* __restrict__ ssum,
                                      const float* __restrict__ h, float* __restrict__ out,
                                      int H, int C) {
    const int HC = H * C;
    const int lane = threadIdx.x & 31;
    const int wave = threadIdx.x >> 5;
    long long e = (long long)blockIdx.x * (blockDim.x >> 5) + wave;
    if (e >= E + N) return;
    int s = (e < E) ? esrc[e] : (int)(e - E);
    int d = (e < E) ? edst[e] : (int)(e - E);
    const int cpl = HC >> 5;                       // channels per lane (4 for 128ch, 2 for 64ch)
    for (int j = 0; j < cpl; ++j) {
        int ch = lane * cpl + j;
        int hh = ch / C;
        float ee = as[(size_t)s * H + hh] + ad[(size_t)d * H + hh];
        ee = ee > 0.f ? ee : NEG_ATT * ee;
        float alpha = expf(ee - fdec(m[(size_t)d * H + hh])) /
                      (ssum[(size_t)d * H + hh] + 1e-16f);
        atomicAdd(&out[(size_t)d * HC + ch], alpha * h[(size_t)s * HC + ch]);
    }
}

// ---------- bias (+ optional leaky-relu), in place ----------
__global__ void bias_act_kernel(float* __restrict__ x, const float* __restrict__ b,
                                long long total, int HC, float slope, int doAct) {
    long long i = (long long)blockIdx.x * blockDim.x + threadIdx.x;
    if (i >= total) return;
    float v = x[i] + b[i % HC];
    if (doAct) v = v > 0.f ? v : slope * v;
    x[i] = v;
}

// ---------- fused decoder: pred = relu([z[s]||z[d]] @ Wd1 + bd1) @ Wd2 + bd2 ----------
// wave = 16 edges; A-frag gathered on the fly; two WMMA accs cover 32 hidden units.
__global__ void decoder_wmma_kernel(const float* __restrict__ z,
                                    const int* __restrict__ esrc, const int* __restrict__ edst,
                                    const float* __restrict__ Wd1, const float* __restrict__ bd1,
                                    const float* __restrict__ Wd2, const float* __restrict__ bd2,
                                    float* __restrict__ pred, long long E, int Dz, int Hd) {
    const int lane = threadIdx.x & 31;
    const int wave = threadIdx.x >> 5;
    long long tile = (long long)blockIdx.x * (blockDim.x >> 5) + wave;
    long long tiles = (E + 15) >> 4;
    if (tile >= tiles) return;                     // wave-uniform
    const long long e0 = tile << 4;

    long long er = e0 + (lane & 15); if (er >= E) er = E - 1;
    const int snode = esrc[er];
    const int dnode = edst[er];
    const int kp = (lane >> 4) << 1;
    const int ncol = lane & 15;
    const int K = 2 * Dz;                          // 128

    v8f acc0 = {}, acc1 = {};
    for (int k0 = 0; k0 < K; k0 += 4) {
        int k = k0 + kp;                           // k,k+1 stay within one half (Dz % 4 == 0)
        v2f a, b0, b1;
        const float* zp = (k < Dz) ? (z + (size_t)snode * Dz + k)
                                   : (z + (size_t)dnode * Dz + (k - Dz));
        a[0] = zp[0]; a[1] = zp[1];
        b0[0] = Wd1[(size_t)k * Hd + ncol];
        b0[1] = Wd1[(size_t)(k + 1) * Hd + ncol];
        b1[0] = Wd1[(size_t)k * Hd + 16 + ncol];
        b1[1] = Wd1[(size_t)(k + 1) * Hd + 16 + ncol];
        acc0 = __builtin_amdgcn_wmma_f32_16x16x4_f32(false, a, false, b0,
                                                     (short)0, acc0, false, false);
        acc1 = __builtin_amdgcn_wmma_f32_16x16x4_f32(false, a, false, b1,
                                                     (short)0, acc1, false, false);
    }
    // epilogue: bias, relu, dot with Wd2, reduce across 16 lanes
    const float w20 = Wd2[ncol], w21 = Wd2[16 + ncol];
    const float b10 = bd1[ncol], b11 = bd1[16 + ncol];
    float part[8];
    for (int g = 0; g < 8; ++g) {
        float h0 = acc0[g] + b10; h0 = h0 > 0.f ? h0 : 0.f;
        float h1 = acc1[g] + b11; h1 = h1 > 0.f ? h1 : 0.f;
        part[g] = h0 * w20 + h1 * w21;
    }
    for (int mask = 1; mask < 16; mask <<= 1)
        for (int g = 0; g < 8; ++g)
            part[g] += __shfl_xor(part[g], mask, 32);
    if (ncol == 0) {
        const float b2v = bd2[0];
        long long mbase = e0 + ((lane >> 4) << 3);
        for (int g = 0; g < 8; ++g) {
            long long r = mbase + g;
            if (r < E) pred[r] = part[g] + b2v;
        }
    }
}

// ---------- host ----------
static inline unsigned grid1(long long n, int bs) { return (unsigned)((n + bs - 1) / bs); }

extern "C" void kernel_launch(void* const* d_in, const int* in_sizes, int n_in,
                              void* d_out, int out_size, void* d_ws, size_t ws_size,
                              hipStream_t stream) {
    const float* x     = (const float*)d_in[0];
    const int*   ei    = (const int*)d_in[1];
    const float* W1    = (const float*)d_in[2];
    const float* a1s_w = (const float*)d_in[3];
    const float* a1d_w = (const float*)d_in[4];
    const float* b1    = (const float*)d_in[5];
    const float* W2    = (const float*)d_in[6];
    const float* a2s_w = (const float*)d_in[7];
    const float* a2d_w = (const float*)d_in[8];
    const float* b2    = (const float*)d_in[9];
    const float* Wd1   = (const float*)d_in[10];
    const float* bd1   = (const float*)d_in[11];
    const float* Wd2   = (const float*)d_in[12];
    const float* bd2   = (const float*)d_in[13];

    const int H1  = 4;
    const int HC1 = in_sizes[3];             // 128
    const int C1  = HC1 / H1;                // 32
    const int in_dim = in_sizes[2] / HC1;    // 128
    const int N   = in_sizes[0] / in_dim;    // 50000
    const long long E = in_sizes[1] / 2;     // 1,600,000
    const int Dz  = in_sizes[9];             // 64
    const int Hd  = in_sizes[11];            // 32

    const int* esrc = ei;
    const int* edst = ei + E;

    // workspace layout
    float*    h1   = (float*)d_ws;
    float*    out1 = h1 + (size_t)N * HC1;
    float*    h2   = out1 + (size_t)N * HC1;
    float*    as1  = h2 + (size_t)N * Dz;
    float*    ad1  = as1 + (size_t)N * H1;
    unsigned* m1   = (unsigned*)(ad1 + (size_t)N * H1);
    float*    s1   = (float*)(m1 + (size_t)N * H1);
    float*    as2  = s1 + (size_t)N * H1;
    float*    ad2  = as2 + N;
    unsigned* m2   = (unsigned*)(ad2 + N);
    float*    s2   = (float*)(m2 + N);

    float* pred = (float*)d_out;             // [E]
    float* z    = pred + E;                  // [N, Dz]

    const int BS = 256;
    const long long EN = E + N;

    // --- init accumulators (poisoned buffers -> zero what we accumulate into) ---
    fill_u32_kernel<<<grid1((long long)N * HC1, BS), BS, 0, stream>>>((unsigned*)out1, 0u, (long long)N * HC1);
    fill_u32_kernel<<<grid1((long long)N * H1,  BS), BS, 0, stream>>>(m1, 0u, (long long)N * H1);
    fill_u32_kernel<<<grid1((long long)N * H1,  BS), BS, 0, stream>>>((unsigned*)s1, 0u, (long long)N * H1);
    fill_u32_kernel<<<grid1((long long)N * Dz,  BS), BS, 0, stream>>>((unsigned*)z, 0u, (long long)N * Dz);
    fill_u32_kernel<<<grid1(N, BS), BS, 0, stream>>>(m2, 0u, N);
    fill_u32_kernel<<<grid1(N, BS), BS, 0, stream>>>((unsigned*)s2, 0u, N);

    // --- layer 1: h1 = x @ W1 ---
    {
        long long waves = (((long long)N + 15) >> 4) * (HC1 >> 4);
        gemm_wmma_f32<<<grid1(waves, 8), BS, 0, stream>>>(x, W1, h1, N, HC1, in_dim);
    }
    attn_scores_kernel<<<grid1((long long)N * H1, BS), BS, 0, stream>>>(h1, a1s_w, a1d_w, as1, ad1, N, H1, C1);
    edge_max_kernel<<<grid1(EN, BS), BS, 0, stream>>>(esrc, edst, E, N, as1, ad1, m1, H1);
    edge_sum_kernel<<<grid1(EN, BS), BS, 0, stream>>>(esrc, edst, E, N, as1, ad1, m1, s1, H1);
    edge_aggregate_kernel<<<grid1(EN, 8), BS, 0, stream>>>(esrc, edst, E, N, as1, ad1, m1, s1, h1, out1, H1, C1);
    bias_act_kernel<<<grid1((long long)N * HC1, BS), BS, 0, stream>>>(out1, b1, (long long)N * HC1, HC1, NEG_ACT, 1);

    // --- layer 2: h2 = out1 @ W2 ---
    {
        long long waves = (((long long)N + 15) >> 4) * (Dz >> 4);
        gemm_wmma_f32<<<grid1(waves, 8), BS, 0, stream>>>(out1, W2, h2, N, Dz, HC1);
    }
    attn_scores_kernel<<<grid1(N, BS), BS, 0, stream>>>(h2, a2s_w, a2d_w, as2, ad2, N, 1, Dz);
    edge_max_kernel<<<grid1(EN, BS), BS, 0, stream>>>(esrc, edst, E, N, as2, ad2, m2, 1);
    edge_sum_kernel<<<grid1(EN, BS), BS, 0, stream>>>(esrc, edst, E, N, as2, ad2, m2, s2, 1);
    edge_aggregate_kernel<<<grid1(EN, 8), BS, 0, stream>>>(esrc, edst, E, N, as2, ad2, m2, s2, h2, z, 1, Dz);
    bias_act_kernel<<<grid1((long long)N * Dz, BS), BS, 0, stream>>>(z, b2, (long long)N * Dz, Dz, 0.f, 0);

    // --- decoder over original edges ---
    {
        long long tiles = (E + 15) >> 4;
        decoder_wmma_kernel<<<grid1(tiles, 8), BS, 0, stream>>>(z, esrc, edst, Wd1, bd1, Wd2, bd2, pred, E, Dz, Hd);
    }
}